// TransformerEncoder_34643206209942
// MI455X (gfx1250) — compile-verified
//
#include <hip/hip_runtime.h>
#include <math.h>

// ---------------------------------------------------------------------------
// T5-style transformer encoder for MI455X (gfx1250), fp32 WMMA path with
// async Global->LDS double-buffered staging (CDNA5 ASYNCcnt engine).
// B=2, L=1024, D=1024, F=4096, NL=4, NH=16, DK=64
// ---------------------------------------------------------------------------

typedef __attribute__((ext_vector_type(2))) float v2f;
typedef __attribute__((ext_vector_type(8))) float v8f;

static constexpr int kB  = 2;
static constexpr int kL  = 1024;
static constexpr int kD  = 1024;
static constexpr int kF  = 4096;
static constexpr int kNL = 4;
static constexpr int kNH = 16;
static constexpr int kDK = 64;
static constexpr int kHD = kNH * kDK;   // 1024 == kD
static constexpr int kQB = 32;          // query rows per attention block
static constexpr int kKS = 68;          // Q/K LDS row stride (16B aligned, bank-safe)

#define WMMA_F32(a, b, c) \
    __builtin_amdgcn_wmma_f32_16x16x4_f32(false, (a), false, (b), (short)0, (c), false, false)

// Async Global->LDS copy, 16 bytes per lane, tracked by ASYNCcnt.
__device__ __forceinline__ void async_copy_b128(const void* lds_ptr, const void* gptr) {
    unsigned ldsOff = (unsigned)(unsigned long long)lds_ptr;   // low 32b of generic = LDS offset
    unsigned long long ga = (unsigned long long)gptr;
    asm volatile("global_load_async_to_lds_b128 %0, %1, off"
                 :: "v"(ldsOff), "v"(ga)
                 : "memory");
}
__device__ __forceinline__ void wait_async0() {
    asm volatile("s_wait_asynccnt 0x0" ::: "memory");
}

// ---------------------------------------------------------------------------
// Kernel 1: T5 relative-position bucket table  bucket[l*L + s], rel = s - l
// ---------------------------------------------------------------------------
__global__ __launch_bounds__(256) void bucket_kernel(int* __restrict__ bucket) {
    int idx = blockIdx.x * 256 + threadIdx.x;          // over L*L
    int l = idx >> 10, s = idx & (kL - 1);
    int rp = s - l;
    int rb = (rp > 0) ? 16 : 0;                        // NUM_BUCKETS/2
    int arp = rp < 0 ? -rp : rp;
    int val;
    if (arp < 8) {                                     // max_exact = 8
        val = arp;
    } else {
        const float scale = 8.0f / logf(16.0f);        // (16-8)/log(128/8)
        int large = 8 + (int)(logf((float)arp * 0.125f) * scale);
        val = large < 15 ? large : 15;
    }
    bucket[idx] = rb + val;
}

// ---------------------------------------------------------------------------
// Kernel 2: RMSNorm, one 256-thread block per row of D=1024
// ---------------------------------------------------------------------------
__global__ __launch_bounds__(256) void rmsnorm_kernel(const float* __restrict__ x,
                                                      const float* __restrict__ w,
                                                      float* __restrict__ y) {
    __shared__ float red[8];
    int row = blockIdx.x, tid = threadIdx.x;
    const float* xr = x + (size_t)row * kD;
    float* yr = y + (size_t)row * kD;
    float ss = 0.f;
    for (int i = tid; i < kD; i += 256) { float v = xr[i]; ss += v * v; }
    for (int off = 16; off > 0; off >>= 1) ss += __shfl_xor(ss, off, 32);
    if ((tid & 31) == 0) red[tid >> 5] = ss;
    __syncthreads();
    if (tid == 0) {
        float t = 0.f;
        for (int i = 0; i < 8; ++i) t += red[i];
        red[0] = rsqrtf(t * (1.0f / (float)kD) + 1e-6f);
    }
    __syncthreads();
    float scale = red[0];
    for (int i = tid; i < kD; i += 256) yr[i] = xr[i] * scale * w[i];
}

// ---------------------------------------------------------------------------
// Kernel 3: fp32 WMMA GEMM  C[M,N] = act(A[M,K] @ W[K,N]) (+ resid)
// Block tile 64x128, 8 waves, each wave a 32x32 tile (4x v_wmma_f32_16x16x4).
// Double-buffered LDS tiles filled by async Global->LDS b128 copies.
// act: 0 = none, 1 = relu.  resid != nullptr -> C = resid + A@W.
// ---------------------------------------------------------------------------
__device__ __forceinline__ void gemm_store(float* C, const float* resid, int act,
                                           size_t idx, float v) {
    if (act) v = fmaxf(v, 0.0f);
    if (resid) v += resid[idx];
    C[idx] = v;
}

__global__ __launch_bounds__(256) void gemm_kernel(const float* __restrict__ A,
                                                   const float* __restrict__ W,
                                                   float* __restrict__ C,
                                                   const float* __restrict__ resid,
                                                   int M, int N, int K, int act) {
    constexpr int BM = 64, BN = 128, BK = 32;
    __shared__ __align__(16) float As[2][BM][BK + 4];   // row stride 36 floats (144B)
    __shared__ __align__(16) float Bs[2][BK][BN + 4];   // row stride 132 floats (528B)

    int tid = threadIdx.x;
    int wave = tid >> 5, lane = tid & 31;
    int wm = (wave >> 2) * 32;        // 0 / 32
    int wn = (wave & 3) * 32;         // 0..96
    int lh = lane & 15;               // lane within half-wave
    int kq = (lane >> 4) * 2;         // K sub-offset per WMMA f32 A/B layout
    int rowHalf = (lane >> 4) * 8;    // C layout: upper half-wave holds M+8

    int blockM = blockIdx.y * BM;
    int blockN = blockIdx.x * BN;

    // Per-thread staging coordinates: A tile = 512 float4 (2/thread),
    // B tile = 1024 float4 (4/thread).
    int aRow0 = tid >> 3,          aCol0 = (tid & 7) * 4;
    int aRow1 = (tid + 256) >> 3,  aCol1 = ((tid + 256) & 7) * 4;

    auto stage = [&](int st, int k0) {
        async_copy_b128(&As[st][aRow0][aCol0], A + (size_t)(blockM + aRow0) * K + k0 + aCol0);
        async_copy_b128(&As[st][aRow1][aCol1], A + (size_t)(blockM + aRow1) * K + k0 + aCol1);
#pragma unroll
        for (int r = 0; r < 4; ++r) {
            int idx = tid + r * 256;
            int row = idx >> 5, col4 = (idx & 31) * 4;
            async_copy_b128(&Bs[st][row][col4], W + (size_t)(k0 + row) * N + blockN + col4);
        }
    };

    v8f acc00 = {}, acc01 = {}, acc10 = {}, acc11 = {};

    stage(0, 0);
    wait_async0();
    __syncthreads();

    int nK = K / BK;
    for (int t = 0; t < nK; ++t) {
        int st = t & 1;
        if (t + 1 < nK) stage(st ^ 1, (t + 1) * BK);

#pragma unroll
        for (int kk = 0; kk < BK; kk += 4) {
            v2f a0, a1, b0, b1;
            a0.x = As[st][wm + lh][kk + kq];          a0.y = As[st][wm + lh][kk + kq + 1];
            a1.x = As[st][wm + 16 + lh][kk + kq];     a1.y = As[st][wm + 16 + lh][kk + kq + 1];
            b0.x = Bs[st][kk + kq][wn + lh];          b0.y = Bs[st][kk + kq + 1][wn + lh];
            b1.x = Bs[st][kk + kq][wn + 16 + lh];     b1.y = Bs[st][kk + kq + 1][wn + 16 + lh];
            acc00 = WMMA_F32(a0, b0, acc00);
            acc01 = WMMA_F32(a0, b1, acc01);
            acc10 = WMMA_F32(a1, b0, acc10);
            acc11 = WMMA_F32(a1, b1, acc11);
        }
        wait_async0();       // next-stage tiles have landed in LDS
        __syncthreads();     // all waves done reading current stage
    }

    int c0 = blockN + wn + lh;
    int c1 = c0 + 16;
#pragma unroll
    for (int i = 0; i < 8; ++i) {
        int r0 = blockM + wm + rowHalf + i;
        int r1 = r0 + 16;
        gemm_store(C, resid, act, (size_t)r0 * N + c0, acc00[i]);
        gemm_store(C, resid, act, (size_t)r0 * N + c1, acc01[i]);
        gemm_store(C, resid, act, (size_t)r1 * N + c0, acc10[i]);
        gemm_store(C, resid, act, (size_t)r1 * N + c1, acc11[i]);
    }
}

// ---------------------------------------------------------------------------
// Kernel 4: attention for one (b, h, 32-query-row block).
// Full score rows S[32][1025] kept in LDS (320KB WGP LDS makes this cheap).
// Pass A: S = Q*K^T + bias (WMMA, K tiles async double-buffered).
// Pass B: exact f32 softmax. Pass C: O = P*V (WMMA, V streamed from L2).
// ---------------------------------------------------------------------------
__global__ __launch_bounds__(256) void attn_kernel(const float* __restrict__ q,
                                                   const float* __restrict__ k,
                                                   const float* __restrict__ v,
                                                   const int* __restrict__ bucket,
                                                   const float* __restrict__ rel_emb,
                                                   float* __restrict__ o) {
    extern __shared__ float smem[];
    float* S    = smem;                          // kQB * (kL + 1)
    float* Qs   = S + kQB * (kL + 1);            // kQB * kKS
    float* Ks   = Qs + kQB * kKS;                // 2 * 64 * kKS (double buffer)
    float* relh = Ks + 2 * 64 * kKS;             // 32

    int qb = blockIdx.x * kQB;
    int h  = blockIdx.y;
    int b  = blockIdx.z;
    int tid = threadIdx.x, wave = tid >> 5, lane = tid & 31;
    int lh = lane & 15, kq = (lane >> 4) * 2, rowHalf = (lane >> 4) * 8;
    int wm2 = (wave >> 2) * 16;       // 0 / 16   (query rows)
    int wn2 = (wave & 3) * 16;        // 0..48    (key cols / d cols)

    if (tid < 32) relh[tid] = rel_emb[tid * kNH + h];

    const float* qbase = q + ((size_t)(b * kL + qb)) * kHD + h * kDK;
    const float* kbase = k + ((size_t)(b * kL)) * kHD + h * kDK;

    // Async-stage Q block [kQB x kDK]: 512 float4, 2 per thread.
#pragma unroll
    for (int r = 0; r < 2; ++r) {
        int idx = tid + r * 256;
        int row = idx >> 4, col4 = (idx & 15) * 4;
        async_copy_b128(&Qs[row * kKS + col4], qbase + (size_t)row * kHD + col4);
    }
    // K-tile staging: 64x64 = 1024 float4, 4 per thread.
    auto stageK = [&](int st, int s0) {
#pragma unroll
        for (int r = 0; r < 4; ++r) {
            int idx = tid + r * 256;
            int row = idx >> 4, col4 = (idx & 15) * 4;
            async_copy_b128(&Ks[(size_t)st * 64 * kKS + row * kKS + col4],
                            kbase + (size_t)(s0 + row) * kHD + col4);
        }
    };

    stageK(0, 0);
    wait_async0();
    __syncthreads();

    // ---- Pass A: scores + bias --------------------------------------------
    for (int t = 0; t < kL / 64; ++t) {
        int st = t & 1;
        int s0 = t * 64;
        if (t + 1 < kL / 64) stageK(st ^ 1, s0 + 64);

        const float* Kst = Ks + (size_t)st * 64 * kKS;
        v8f acc = {};
#pragma unroll
        for (int kk = 0; kk < kDK; kk += 4) {
            v2f a, bb;
            a.x  = Qs[(wm2 + lh) * kKS + kk + kq];
            a.y  = Qs[(wm2 + lh) * kKS + kk + kq + 1];
            bb.x = Kst[(wn2 + lh) * kKS + kk + kq];      // B[d][s] = K[s][d]
            bb.y = Kst[(wn2 + lh) * kKS + kk + kq + 1];
            acc = WMMA_F32(a, bb, acc);
        }
#pragma unroll
        for (int i = 0; i < 8; ++i) {
            int r = wm2 + rowHalf + i;        // query row within block
            int c = s0 + wn2 + lh;            // global key index
            float bias = relh[bucket[(size_t)(qb + r) * kL + c]];
            S[r * (kL + 1) + c] = acc[i] + bias;
        }
        wait_async0();
        __syncthreads();
    }

    // ---- Pass B: row softmax (exact, f32) ---------------------------------
    for (int r = wave * 4; r < wave * 4 + 4; ++r) {
        float* Sr = S + r * (kL + 1);
        float m = -3.402823466e38f;
        for (int c = lane; c < kL; c += 32) m = fmaxf(m, Sr[c]);
        for (int off = 16; off > 0; off >>= 1) m = fmaxf(m, __shfl_xor(m, off, 32));
        float sum = 0.f;
        for (int c = lane; c < kL; c += 32) { float e = __expf(Sr[c] - m); Sr[c] = e; sum += e; }
        for (int off = 16; off > 0; off >>= 1) sum += __shfl_xor(sum, off, 32);
        float inv = 1.0f / sum;
        for (int c = lane; c < kL; c += 32) Sr[c] *= inv;
    }
    __syncthreads();

    // ---- Pass C: O = P @ V -------------------------------------------------
    v8f oacc = {};
    const float* vbase = v + ((size_t)(b * kL)) * kHD + h * kDK + wn2 + lh;
    for (int kk = 0; kk < kL; kk += 4) {
        v2f a, bb;
        a.x  = S[(wm2 + lh) * (kL + 1) + kk + kq];
        a.y  = S[(wm2 + lh) * (kL + 1) + kk + kq + 1];
        bb.x = vbase[(size_t)(kk + kq) * kHD];            // B[s][d] = V[s][d]
        bb.y = vbase[(size_t)(kk + kq + 1) * kHD];
        oacc = WMMA_F32(a, bb, oacc);
    }
    float* obase = o + ((size_t)(b * kL + qb)) * kHD + h * kDK;
#pragma unroll
    for (int i = 0; i < 8; ++i) {
        int r = wm2 + rowHalf + i;
        int c = wn2 + lh;
        obase[(size_t)r * kHD + c] = oacc[i];
    }
}

// ---------------------------------------------------------------------------
// Host launcher
// ---------------------------------------------------------------------------
extern "C" void kernel_launch(void* const* d_in, const int* in_sizes, int n_in,
                              void* d_out, int out_size, void* d_ws, size_t ws_size,
                              hipStream_t stream) {
    (void)in_sizes; (void)n_in; (void)out_size; (void)ws_size;

    const float* x       = (const float*)d_in[0];
    const float* rel_emb = (const float*)d_in[1];
    const float* Wq      = (const float*)d_in[2];
    const float* Wk      = (const float*)d_in[3];
    const float* Wv      = (const float*)d_in[4];
    const float* Wo      = (const float*)d_in[5];
    const float* ln1_w   = (const float*)d_in[6];
    const float* ln2_w   = (const float*)d_in[7];
    const float* wi      = (const float*)d_in[8];
    const float* wo      = (const float*)d_in[9];
    const float* ln_f    = (const float*)d_in[10];

    const size_t BL = (size_t)kB * kL;   // 2048 rows
    float* ws   = (float*)d_ws;
    float* xcur = ws;  ws += BL * kD;
    float* y    = ws;  ws += BL * kD;
    float* qb   = ws;  ws += BL * kD;
    float* kb   = ws;  ws += BL * kD;
    float* vb   = ws;  ws += BL * kD;
    float* ob   = ws;  ws += BL * kD;
    float* hb   = ws;  ws += BL * kF;
    int* bucket = (int*)ws;              // kL*kL ints

    hipMemcpyAsync(xcur, x, BL * kD * sizeof(float), hipMemcpyDeviceToDevice, stream);
    bucket_kernel<<<(kL * kL) / 256, 256, 0, stream>>>(bucket);

    const dim3 gProj(kD / 128, (int)BL / 64);   // N=1024 GEMMs
    const dim3 gFF1(kF / 128, (int)BL / 64);    // N=4096 GEMM
    const dim3 gAttn(kL / kQB, kNH, kB);
    const size_t attnLds =
        (size_t)(kQB * (kL + 1) + kQB * kKS + 2 * 64 * kKS + 32) * sizeof(float);
    hipFuncSetAttribute((const void*)attn_kernel,
                        hipFuncAttributeMaxDynamicSharedMemorySize, (int)attnLds);

    for (int l = 0; l < kNL; ++l) {
        const float* Wql = Wq + (size_t)l * kD * kHD;
        const float* Wkl = Wk + (size_t)l * kD * kHD;
        const float* Wvl = Wv + (size_t)l * kD * kHD;
        const float* Wol = Wo + (size_t)l * kHD * kD;
        const float* wil = wi + (size_t)l * kD * kF;
        const float* wol = wo + (size_t)l * kF * kD;

        rmsnorm_kernel<<<(int)BL, 256, 0, stream>>>(xcur, ln1_w + (size_t)l * kD, y);
        gemm_kernel<<<gProj, 256, 0, stream>>>(y, Wql, qb, nullptr, (int)BL, kD, kD, 0);
        gemm_kernel<<<gProj, 256, 0, stream>>>(y, Wkl, kb, nullptr, (int)BL, kD, kD, 0);
        gemm_kernel<<<gProj, 256, 0, stream>>>(y, Wvl, vb, nullptr, (int)BL, kD, kD, 0);
        attn_kernel<<<gAttn, 256, attnLds, stream>>>(qb, kb, vb, bucket, rel_emb, ob);
        gemm_kernel<<<gProj, 256, 0, stream>>>(ob, Wol, xcur, xcur, (int)BL, kD, kD, 0);

        rmsnorm_kernel<<<(int)BL, 256, 0, stream>>>(xcur, ln2_w + (size_t)l * kD, y);
        gemm_kernel<<<gFF1, 256, 0, stream>>>(y, wil, hb, nullptr, (int)BL, kF, kD, 1);
        gemm_kernel<<<gProj, 256, 0, stream>>>(hb, wol, xcur, xcur, (int)BL, kD, kF, 0);
    }
    rmsnorm_kernel<<<(int)BL, 256, 0, stream>>>(xcur, ln_f, (float*)d_out);
}